// RNN_23922967839275
// MI455X (gfx1250) — compile-verified
//
#include <hip/hip_runtime.h>

typedef float v2f __attribute__((ext_vector_type(2)));
typedef float v4f __attribute__((ext_vector_type(4)));
typedef float v8f __attribute__((ext_vector_type(8)));

#define WAVES_PER_BLOCK 8
#define ROWS_PER_WAVE   32                                  // 2 x 16-row WMMA tiles
#define ROWS_PER_BLOCK  (WAVES_PER_BLOCK * ROWS_PER_WAVE)   // 256

// ---------------------------------------------------------------------------
// Dual-matrix GEMV via V_WMMA_F32_16X16X4_F32 (wave32).
//   out[r] = W1[r,:]*v1 + (K2? W2[r,:]*v2 : 0) + b1[r] + (b2? b2[r] : 0)
//
// Each wave owns TWO 16-row tiles (independent accumulator chains for matrix-
// pipe ILP; B fragment reused for both). Per K-step of 8:
//   - one global_load_b128 per tile: lane l<16 gets W[r0+l][k0..k0+3],
//     lane l>=16 gets W[r0+l-16][k0+4..k0+7]  (lanes m / m+16 co-stream row m)
//   - one ds_load_b128 of the LDS-staged vector with the same lane->k mapping
//   - two WMMAs consume the lo/hi b64 halves. Correct for any consistent HW
//     fragment layout because A and B source k identically per (half, VGPR),
//     and GEMV K-order is permutation-invariant.
// Weights are 646 MB streamed once -> non-temporal loads (spare the 192MB L2).
// All 16 columns of D equal the 16-row partials; lanes 0/16 write out.
// ---------------------------------------------------------------------------
__global__ __launch_bounds__(256) void gemv_wmma_kernel(
    const float* __restrict__ W1, int K1, const float* __restrict__ v1,
    const float* __restrict__ W2, int K2, const float* __restrict__ v2,
    const float* __restrict__ b1, const float* __restrict__ b2,
    float* __restrict__ out, int R)
{
    __shared__ float sv[4096];                    // up to K1+K2 = 4096 floats (16 KB)

    const int tid = threadIdx.x;
    for (int i = tid; i < K1; i += 256) sv[i] = v1[i];
    if (K2 > 0) {
        for (int i = tid; i < K2; i += 256) sv[K1 + i] = v2[i];
    }
    __syncthreads();

    const int lane    = tid & 31;
    const int wave    = tid >> 5;
    const int rowBase = (blockIdx.x * WAVES_PER_BLOCK + wave) * ROWS_PER_WAVE;
    if (rowBase >= R) return;                     // wave-uniform exit (EXEC all-1 for WMMA)

    const int mrow = lane & 15;                   // row within tile
    const int koff = (lane >> 4) * 4;             // K sub-offset: 0 (lanes 0-15) / 4 (lanes 16-31)
    const int r0   = min(rowBase + mrow,      R - 1);   // clamp tail rows (stores guarded)
    const int r1   = min(rowBase + 16 + mrow, R - 1);

    v8f acc0 = {0.f, 0.f, 0.f, 0.f, 0.f, 0.f, 0.f, 0.f};
    v8f acc1 = {0.f, 0.f, 0.f, 0.f, 0.f, 0.f, 0.f, 0.f};

    {
        const float* w0 = W1 + (size_t)r0 * (size_t)K1 + koff;
        const float* w1 = W1 + (size_t)r1 * (size_t)K1 + koff;
        #pragma unroll 2
        for (int k0 = 0; k0 < K1; k0 += 8) {
            v4f a0 = __builtin_nontemporal_load((const v4f*)(w0 + k0));
            v4f a1 = __builtin_nontemporal_load((const v4f*)(w1 + k0));
            v4f b  = *(const v4f*)(&sv[k0 + koff]);
            acc0 = __builtin_amdgcn_wmma_f32_16x16x4_f32(false, a0.lo, false, b.lo, (short)0, acc0, false, false);
            acc1 = __builtin_amdgcn_wmma_f32_16x16x4_f32(false, a1.lo, false, b.lo, (short)0, acc1, false, false);
            acc0 = __builtin_amdgcn_wmma_f32_16x16x4_f32(false, a0.hi, false, b.hi, (short)0, acc0, false, false);
            acc1 = __builtin_amdgcn_wmma_f32_16x16x4_f32(false, a1.hi, false, b.hi, (short)0, acc1, false, false);
        }
    }
    if (K2 > 0) {
        const float* w0 = W2 + (size_t)r0 * (size_t)K2 + koff;
        const float* w1 = W2 + (size_t)r1 * (size_t)K2 + koff;
        const float* sv2 = sv + K1;
        #pragma unroll 2
        for (int k0 = 0; k0 < K2; k0 += 8) {
            v4f a0 = __builtin_nontemporal_load((const v4f*)(w0 + k0));
            v4f a1 = __builtin_nontemporal_load((const v4f*)(w1 + k0));
            v4f b  = *(const v4f*)(&sv2[k0 + koff]);
            acc0 = __builtin_amdgcn_wmma_f32_16x16x4_f32(false, a0.lo, false, b.lo, (short)0, acc0, false, false);
            acc1 = __builtin_amdgcn_wmma_f32_16x16x4_f32(false, a1.lo, false, b.lo, (short)0, acc1, false, false);
            acc0 = __builtin_amdgcn_wmma_f32_16x16x4_f32(false, a0.hi, false, b.hi, (short)0, acc0, false, false);
            acc1 = __builtin_amdgcn_wmma_f32_16x16x4_f32(false, a1.hi, false, b.hi, (short)0, acc1, false, false);
        }
    }

    // D layout (16x16 f32 C/D): VGPR j of lane n holds M = j + 8*(n>=16).
    // All columns identical -> lane 0 has rows 0..7, lane 16 has rows 8..15.
    if ((lane & 15) == 0) {
        const int mbase = (lane >> 4) * 8;
        #pragma unroll
        for (int j = 0; j < 8; ++j) {
            const int ra = rowBase + mbase + j;          // tile 0
            if (ra < R) {
                float val = acc0[j];
                if (b1) val += b1[ra];
                if (b2) val += b2[ra];
                out[ra] = val;
            }
            const int rb = rowBase + 16 + mbase + j;     // tile 1
            if (rb < R) {
                float val = acc1[j];
                if (b1) val += b1[rb];
                if (b2) val += b2[rb];
                out[rb] = val;
            }
        }
    }
}

// ---------------------------------------------------------------------------
// LSTM pointwise: gates = [i | f | g | o] (PyTorch order), h = o*tanh(f*c+i*g)
// ---------------------------------------------------------------------------
__global__ void lstm_act_kernel(const float* __restrict__ gates,
                                const float* __restrict__ c,
                                float* __restrict__ h_out, int H)
{
    const int i = blockIdx.x * blockDim.x + threadIdx.x;
    if (i >= H) return;
    const float gi = gates[i];
    const float gf = gates[i + H];
    const float gg = gates[i + 2 * H];
    const float go = gates[i + 3 * H];
    const float si = 1.f / (1.f + expf(-gi));
    const float sf = 1.f / (1.f + expf(-gf));
    const float tg = tanhf(gg);
    const float so = 1.f / (1.f + expf(-go));
    const float cn = sf * c[i] + si * tg;
    h_out[i] = so * tanhf(cn);
}

// ---------------------------------------------------------------------------
// Log-softmax: online (max, sumexp) partials -> combine -> subtract logZ
// ---------------------------------------------------------------------------
__global__ __launch_bounds__(256) void lse_partial_kernel(
    const float* __restrict__ x, int n,
    float* __restrict__ pm, float* __restrict__ ps)
{
    __shared__ float sm[256], ss[256];
    const int tid = threadIdx.x;
    const int stride = gridDim.x * 256;

    float m = -__builtin_inff();
    float s = 0.f;
    for (int i = blockIdx.x * 256 + tid; i < n; i += stride) {
        const float v = x[i];
        if (v > m) { s *= expf(m - v); m = v; }   // exp(-inf)=0 on first hit
        s += expf(v - m);
    }
    sm[tid] = m; ss[tid] = s;
    __syncthreads();
    for (int off = 128; off > 0; off >>= 1) {
        if (tid < off) {
            const float m1 = sm[tid], s1 = ss[tid];
            const float m2 = sm[tid + off], s2 = ss[tid + off];
            const float M = fmaxf(m1, m2);
            float S = 0.f;
            if (m1 > -__builtin_inff()) S += s1 * expf(m1 - M);
            if (m2 > -__builtin_inff()) S += s2 * expf(m2 - M);
            sm[tid] = M; ss[tid] = S;
        }
        __syncthreads();
    }
    if (tid == 0) { pm[blockIdx.x] = sm[0]; ps[blockIdx.x] = ss[0]; }
}

__global__ __launch_bounds__(256) void lse_combine_kernel(
    const float* __restrict__ pm, const float* __restrict__ ps, int nb,
    float* __restrict__ logZ)
{
    __shared__ float sm[256], ss[256];
    const int tid = threadIdx.x;
    sm[tid] = (tid < nb) ? pm[tid] : -__builtin_inff();
    ss[tid] = (tid < nb) ? ps[tid] : 0.f;
    __syncthreads();
    for (int off = 128; off > 0; off >>= 1) {
        if (tid < off) {
            const float m1 = sm[tid], s1 = ss[tid];
            const float m2 = sm[tid + off], s2 = ss[tid + off];
            const float M = fmaxf(m1, m2);
            float S = 0.f;
            if (m1 > -__builtin_inff()) S += s1 * expf(m1 - M);
            if (m2 > -__builtin_inff()) S += s2 * expf(m2 - M);
            sm[tid] = M; ss[tid] = S;
        }
        __syncthreads();
    }
    if (tid == 0) *logZ = sm[0] + logf(ss[0]);
}

__global__ void lsm_write_kernel(const float* __restrict__ logits,
                                 const float* __restrict__ logZ,
                                 float* __restrict__ out, int n)
{
    const int i = blockIdx.x * blockDim.x + threadIdx.x;
    if (i < n) out[i] = logits[i] - *logZ;
}

// ---------------------------------------------------------------------------
extern "C" void kernel_launch(void* const* d_in, const int* in_sizes, int n_in,
                              void* d_out, int out_size, void* d_ws, size_t ws_size,
                              hipStream_t stream)
{
    const float* x     = (const float*)d_in[0];
    const float* h1    = (const float*)d_in[1];
    const float* c1    = (const float*)d_in[2];
    const float* h2    = (const float*)d_in[3];
    const float* c2    = (const float*)d_in[4];
    const float* W_ih1 = (const float*)d_in[5];
    const float* W_hh1 = (const float*)d_in[6];
    const float* b_ih1 = (const float*)d_in[7];
    const float* b_hh1 = (const float*)d_in[8];
    const float* W_ih2 = (const float*)d_in[9];
    const float* W_hh2 = (const float*)d_in[10];
    const float* b_ih2 = (const float*)d_in[11];
    const float* b_hh2 = (const float*)d_in[12];
    const float* W_out = (const float*)d_in[13];
    const float* b_out = (const float*)d_in[14];
    float* out = (float*)d_out;

    const int IN = 1024, H = 2048, V = 50257;
    const int R1 = 4 * H;                       // 8192 gate rows

    // workspace layout (floats); ~290 KB total
    float* ws     = (float*)d_ws;
    float* gates1 = ws;                         // 8192
    float* gates2 = ws + 8192;                  // 8192
    float* h1n    = ws + 16384;                 // 2048
    float* h2n    = ws + 18432;                 // 2048
    float* logits = ws + 20480;                 // 50257
    float* pm     = ws + 20480 + 50432;         // 256 (padded start)
    float* psum   = pm + 256;                   // 256
    float* logZ   = psum + 256;                 // 1

    // Layer 1: gates1 = W_ih1@x + W_hh1@h1 + b_ih1 + b_hh1
    gemv_wmma_kernel<<<R1 / ROWS_PER_BLOCK, 256, 0, stream>>>(
        W_ih1, IN, x, W_hh1, H, h1, b_ih1, b_hh1, gates1, R1);
    lstm_act_kernel<<<H / 256, 256, 0, stream>>>(gates1, c1, h1n, H);

    // Layer 2: gates2 = W_ih2@h1n + W_hh2@h2 + b_ih2 + b_hh2
    gemv_wmma_kernel<<<R1 / ROWS_PER_BLOCK, 256, 0, stream>>>(
        W_ih2, H, h1n, W_hh2, H, h2, b_ih2, b_hh2, gates2, R1);
    lstm_act_kernel<<<H / 256, 256, 0, stream>>>(gates2, c2, h2n, H);

    // Output projection: logits = W_out@h2n + b_out   (412 MB stream, the hot loop)
    const int blocksV = (V + ROWS_PER_BLOCK - 1) / ROWS_PER_BLOCK;  // 197
    gemv_wmma_kernel<<<blocksV, 256, 0, stream>>>(
        W_out, H, h2n, nullptr, 0, nullptr, b_out, nullptr, logits, V);

    // log-softmax
    lse_partial_kernel<<<256, 256, 0, stream>>>(logits, V, pm, psum);
    lse_combine_kernel<<<1, 256, 0, stream>>>(pm, psum, 256, logZ);
    lsm_write_kernel<<<(V + 255) / 256, 256, 0, stream>>>(logits, logZ, out, V);
}